// MultiPooling_31361851195618
// MI455X (gfx1250) — compile-verified
//
#include <hip/hip_runtime.h>
#include <math.h>

#define HIDDEN   128
#define N_NODES  1000000
#define N_GRAPHS 4096

#define NWAVES_SEG     16384
#define NODES_PER_WAVE ((N_NODES + NWAVES_SEG - 1) / NWAVES_SEG)   // 62

typedef __attribute__((ext_vector_type(2))) float v2f;
typedef __attribute__((ext_vector_type(4))) float v4f;
typedef __attribute__((ext_vector_type(8))) float v8f;

#define NEG_BIG (-3.402823466e+38f)

// ---------------------------------------------------------------------------
// Kernel 0: init accumulators (must run every call; ws is poisoned once)
// ---------------------------------------------------------------------------
__global__ void mp_init(float* __restrict__ sums, float* __restrict__ maxs,
                        int* __restrict__ counts) {
    int i = blockIdx.x * blockDim.x + threadIdx.x;
    if (i < N_GRAPHS * HIDDEN) {
        sums[i] = 0.0f;
        maxs[i] = NEG_BIG;
    }
    if (i < N_GRAPHS) counts[i] = 0;
}

// ---------------------------------------------------------------------------
// Kernel 1: streaming segment sum/max/count over sorted batch ids.
// One wave per contiguous slice of nodes; lane l owns columns [4l, 4l+4).
// x is streamed once -> non-temporal loads (TH=NT) keep the 192MB L2 free
// for the accumulator atomics. Register accumulation within a run of equal
// graph ids; atomics only at segment boundaries (~2 per wave).
// ---------------------------------------------------------------------------
__device__ __forceinline__ void mp_flush(int seg, int cnt, const v4f& s,
                                         const v4f& mx, int lane,
                                         float* __restrict__ sums,
                                         float* __restrict__ maxs,
                                         int* __restrict__ counts) {
    if (cnt <= 0) return;
    int base = seg * HIDDEN + lane * 4;
    atomicAdd(&sums[base + 0], s.x);
    atomicAdd(&sums[base + 1], s.y);
    atomicAdd(&sums[base + 2], s.z);
    atomicAdd(&sums[base + 3], s.w);
    atomicMax(&maxs[base + 0], mx.x);
    atomicMax(&maxs[base + 1], mx.y);
    atomicMax(&maxs[base + 2], mx.z);
    atomicMax(&maxs[base + 3], mx.w);
    if (lane == 0) atomicAdd(&counts[seg], cnt);
}

__global__ void mp_segreduce(const float* __restrict__ x,
                             const int* __restrict__ batch,
                             float* __restrict__ sums,
                             float* __restrict__ maxs,
                             int* __restrict__ counts) {
    int wave = (blockIdx.x * blockDim.x + threadIdx.x) >> 5;
    int lane = threadIdx.x & 31;
    long long start = (long long)wave * NODES_PER_WAVE;
    if (start >= N_NODES) return;
    long long end = start + NODES_PER_WAVE;
    if (end > N_NODES) end = N_NODES;

    const v4f* xv = (const v4f*)x;   // 32 x v4f per node row
    v4f s  = (v4f){0.f, 0.f, 0.f, 0.f};
    v4f mx = (v4f){NEG_BIG, NEG_BIG, NEG_BIG, NEG_BIG};
    int cnt = 0;
    int cur = -1;

    // prefetch first batch id to break the load->branch->load serial chain
    int g_next = __builtin_nontemporal_load(&batch[start]);

    for (long long n = start; n < end; ++n) {
        int g = g_next;
        if (n + 1 < end) g_next = __builtin_nontemporal_load(&batch[n + 1]);
        // issue the streaming row load early (independent of the flush branch)
        v4f v = __builtin_nontemporal_load(&xv[n * 32 + lane]);
        if (g != cur) {
            mp_flush(cur, cnt, s, mx, lane, sums, maxs, counts);
            cur = g; cnt = 0;
            s  = (v4f){0.f, 0.f, 0.f, 0.f};
            mx = (v4f){NEG_BIG, NEG_BIG, NEG_BIG, NEG_BIG};
        }
        s.x += v.x; s.y += v.y; s.z += v.z; s.w += v.w;
        mx.x = fmaxf(mx.x, v.x); mx.y = fmaxf(mx.y, v.y);
        mx.z = fmaxf(mx.z, v.z); mx.w = fmaxf(mx.w, v.w);
        ++cnt;
    }
    mp_flush(cur, cnt, s, mx, lane, sums, maxs, counts);
}

// ---------------------------------------------------------------------------
// Kernel 2: build pooled = [mean | max(guarded) | sum]  -> [G, 384]
// ---------------------------------------------------------------------------
__global__ void mp_pooled(const float* __restrict__ sums,
                          const float* __restrict__ maxs,
                          const int* __restrict__ counts,
                          float* __restrict__ pooled) {
    int i = blockIdx.x * blockDim.x + threadIdx.x;
    if (i >= N_GRAPHS * HIDDEN) return;
    int g = i / HIDDEN;
    int c = i - g * HIDDEN;
    int cnt = counts[g];
    float s = sums[i];
    float m = (cnt > 0) ? maxs[i] : 0.0f;
    float mean = s / fmaxf((float)cnt, 1.0f);
    float* row = pooled + (long long)g * (3 * HIDDEN);
    row[c]              = mean;
    row[HIDDEN + c]     = m;
    row[2 * HIDDEN + c] = s;
}

// ---------------------------------------------------------------------------
// Kernel 3: h = pooled @ W + b via V_WMMA_F32_16X16X4_F32.
// One wave per 16x16 tile: tiles = (4096/16) x (128/16) = 256 x 8 = 2048.
// A frag (16x4 f32): lanes 0-15 -> {K0,K1}, lanes 16-31 -> {K2,K3} (ISA layout)
// B frag (4x16 f32): mirrored (same koff), N = lane%16.
// C/D: VGPR r = row (tileM*16 + r + (lane>=16 ? 8 : 0)), col N = lane%16.
// No divergence: EXEC all-ones as WMMA requires.
// ---------------------------------------------------------------------------
__global__ void mp_gemm_wmma(const float* __restrict__ pooled,
                             const float* __restrict__ W,
                             const float* __restrict__ bias,
                             float* __restrict__ h) {
    int wave  = (blockIdx.x * blockDim.x + threadIdx.x) >> 5;
    int lane  = threadIdx.x & 31;
    int tileN = wave & 7;        // 8 col tiles
    int tileM = wave >> 3;       // 256 row tiles
    int mloc  = lane & 15;
    int koff  = (lane < 16) ? 0 : 2;
    int m = tileM * 16 + mloc;   // A row this lane carries
    int n = tileN * 16 + mloc;   // B/D col this lane carries

    const float* arow = pooled + (long long)m * (3 * HIDDEN);
    v8f c = {};
    for (int k0 = 0; k0 < 3 * HIDDEN; k0 += 4) {
        v2f a, bm;
        // aligned float2 load of A (k0+koff is even)
        a = *(const v2f*)(arow + k0 + koff);
        bm.x = W[(long long)(k0 + koff)     * HIDDEN + n];
        bm.y = W[(long long)(k0 + koff + 1) * HIDDEN + n];
        c = __builtin_amdgcn_wmma_f32_16x16x4_f32(
                /*neg_a=*/false, a, /*neg_b=*/false, bm,
                /*c_mod=*/(short)0, c, /*reuse_a=*/false, /*reuse_b=*/false);
    }
    float bn = bias[n];
    int rbase = tileM * 16 + ((lane < 16) ? 0 : 8);
#pragma unroll
    for (int r = 0; r < 8; ++r) {
        h[(long long)(rbase + r) * HIDDEN + n] = c[r] + bn;
    }
}

// ---------------------------------------------------------------------------
// Kernel 4: LayerNorm (eps 1e-5) + exact GELU. One wave per row; lane holds
// 4 columns; wave32 __shfl_xor reduction. Output written non-temporally
// (never re-read on device).
// ---------------------------------------------------------------------------
__device__ __forceinline__ float mp_gelu(float v) {
    return 0.5f * v * (1.0f + erff(v * 0.70710678118654752f));
}

__global__ void mp_ln_gelu(const float* __restrict__ h,
                           const float* __restrict__ gamma,
                           const float* __restrict__ beta,
                           float* __restrict__ out) {
    int wave = (blockIdx.x * blockDim.x + threadIdx.x) >> 5;
    int lane = threadIdx.x & 31;
    if (wave >= N_GRAPHS) return;
    v4f v = ((const v4f*)(h + (long long)wave * HIDDEN))[lane];
    float s  = v.x + v.y + v.z + v.w;
    float sq = v.x * v.x + v.y * v.y + v.z * v.z + v.w * v.w;
#pragma unroll
    for (int off = 16; off >= 1; off >>= 1) {
        s  += __shfl_xor(s, off);
        sq += __shfl_xor(sq, off);
    }
    float mu  = s * (1.0f / HIDDEN);
    float var = sq * (1.0f / HIDDEN) - mu * mu;
    float inv = rsqrtf(var + 1e-5f);
    v4f g  = ((const v4f*)gamma)[lane];
    v4f bt = ((const v4f*)beta)[lane];
    v4f o;
    o.x = mp_gelu((v.x - mu) * inv * g.x + bt.x);
    o.y = mp_gelu((v.y - mu) * inv * g.y + bt.y);
    o.z = mp_gelu((v.z - mu) * inv * g.z + bt.z);
    o.w = mp_gelu((v.w - mu) * inv * g.w + bt.w);
    __builtin_nontemporal_store(o, &((v4f*)(out + (long long)wave * HIDDEN))[lane]);
}

// ---------------------------------------------------------------------------
extern "C" void kernel_launch(void* const* d_in, const int* in_sizes, int n_in,
                              void* d_out, int out_size, void* d_ws, size_t ws_size,
                              hipStream_t stream) {
    const float* x     = (const float*)d_in[0];
    const int*   batch = (const int*)d_in[1];
    // d_in[2] = num_graphs scalar (compile-time constant N_GRAPHS)
    const float* W     = (const float*)d_in[3];
    const float* bias  = (const float*)d_in[4];
    const float* gamma = (const float*)d_in[5];
    const float* beta  = (const float*)d_in[6];
    float* out = (float*)d_out;

    // Workspace layout (bytes):
    char* ws = (char*)d_ws;
    float* sums   = (float*)(ws);                                     // 2 MB
    float* maxs   = (float*)(ws + 2097152);                           // 2 MB
    int*   counts = (int*)  (ws + 4194304);                           // 16 KB
    float* pooled = (float*)(ws + 4210688);                           // 6 MB
    float* h      = (float*)(ws + 10502144);                          // 2 MB

    // 0: init accumulators (G*H = 524288 threads)
    mp_init<<<(N_GRAPHS * HIDDEN + 255) / 256, 256, 0, stream>>>(sums, maxs, counts);

    // 1: segment reduce (16384 waves = 2048 blocks x 256 threads)
    mp_segreduce<<<NWAVES_SEG / 8, 256, 0, stream>>>(x, batch, sums, maxs, counts);

    // 2: pooled features
    mp_pooled<<<(N_GRAPHS * HIDDEN + 255) / 256, 256, 0, stream>>>(sums, maxs, counts, pooled);

    // 3: WMMA GEMM + bias (2048 waves = 512 blocks x 128 threads)
    mp_gemm_wmma<<<512, 128, 0, stream>>>(pooled, W, bias, h);

    // 4: LayerNorm + GELU (4096 waves = 512 blocks x 256 threads)
    mp_ln_gelu<<<512, 256, 0, stream>>>(h, gamma, beta, out);
}